// GraphConv_50302656971247
// MI455X (gfx1250) — compile-verified
//
#include <hip/hip_runtime.h>
#include <math.h>

#define N_ENT   150000
#define N_USR   60000
#define CH      64
#define E_EDGES 1000000
#define NNZ_CNT 1000000
#define NRELM1  9        // NREL - 1 valid relation rows
#define NHOPS   2
#define EPS_N   1e-12f

typedef __attribute__((ext_vector_type(2))) float v2f;
typedef __attribute__((ext_vector_type(8))) float v8f;

// ---------------------------------------------------------------------------
// Kernel 1: per-edge attention via WMMA.
// For a tile of 16 edges:  hn2[16] = column-select of (H^2)[16x64] x (R^2)^T[64x16]
// using 16 chained V_WMMA_F32_16X16X4_F32 (exact f32, matches reference math).
// Also scatter segment-max of att into segmax via uint atomicMax (att >= 0).
// ---------------------------------------------------------------------------
#define ATT_WAVES 4
__global__ __launch_bounds__(ATT_WAVES * 32)
void att_wmma_kernel(const float* __restrict__ ent,
                     const float* __restrict__ weight,
                     const int* __restrict__ head,
                     const int* __restrict__ tail,
                     const int* __restrict__ etype,
                     float* __restrict__ att,
                     unsigned int* __restrict__ segmax)
{
    __shared__ float sW2[16 * CH];                  // squared relation table, zero-padded to 16 rows
    __shared__ float sH[ATT_WAVES][16 * CH];        // squared gathered head rows (later reused for C tile)
    __shared__ float sT[ATT_WAVES][16 * CH];        // squared gathered tail rows (later reused for C tile)

    const int lane = threadIdx.x & 31;
    const int wave = threadIdx.x >> 5;

    for (int i = threadIdx.x; i < 16 * CH; i += ATT_WAVES * 32) {
        const int n = i >> 6, c = i & 63;
        const float w = (n < NRELM1) ? weight[n * CH + c] : 0.0f;
        sW2[i] = w * w;
    }
    __syncthreads();

    const int tile = blockIdx.x * ATT_WAVES + wave;
    if (tile * 16 >= E_EDGES) return;               // wave-uniform: EXEC stays all-ones for WMMA
    const int base = tile * 16;

    // Stage 16 head/tail rows, squared; each lane covers 2 channels -> coalesced 256B rows.
    for (int r = 0; r < 16; ++r) {
        const int hr = head[base + r];
        const int tr = tail[base + r];
        const v2f hv = *(const v2f*)(ent + (size_t)hr * CH + 2 * lane);
        const v2f tv = *(const v2f*)(ent + (size_t)tr * CH + 2 * lane);
        sH[wave][r * CH + 2 * lane]     = hv.x * hv.x;
        sH[wave][r * CH + 2 * lane + 1] = hv.y * hv.y;
        sT[wave][r * CH + 2 * lane]     = tv.x * tv.x;
        sT[wave][r * CH + 2 * lane + 1] = tv.y * tv.y;
    }
    // wave-synchronous: same-wave LDS ops are in-order; compiler inserts dscnt waits

    // A layout (16x4 f32): lanes 0-15 -> M=lane, K={4k,4k+1}; lanes 16-31 -> M=lane-16, K={4k+2,4k+3}
    // B layout (4x16 f32): symmetric with N in place of M.
    v8f accH = {};
    v8f accT = {};
    const int m    = lane & 15;
    const int koff = (lane >> 4) << 1;              // 0 or 2
    for (int kk = 0; kk < 16; ++kk) {
        const int c = kk * 4 + koff;
        v2f a, at, b;
        a.x  = sH[wave][m * CH + c];  a.y  = sH[wave][m * CH + c + 1];
        at.x = sT[wave][m * CH + c];  at.y = sT[wave][m * CH + c + 1];
        b.x  = sW2[m * CH + c];       b.y  = sW2[m * CH + c + 1];
        accH = __builtin_amdgcn_wmma_f32_16x16x4_f32(false, a,  false, b, (short)0, accH, false, false);
        accT = __builtin_amdgcn_wmma_f32_16x16x4_f32(false, at, false, b, (short)0, accT, false, false);
    }

    // C layout: VGPR v holds M=v (lanes 0-15, N=lane) and M=v+8 (lanes 16-31, N=lane-16).
    const int mrow = (lane >> 4) << 3;              // 0 or 8
    const int ncol = lane & 15;
#pragma unroll
    for (int v = 0; v < 8; ++v) {
        sH[wave][(mrow + v) * 16 + ncol] = accH[v];
        sT[wave][(mrow + v) * 16 + ncol] = accT[v];
    }

    if (lane < 16) {
        const int e  = base + lane;
        int et = etype[e] - 1;
        if (et < 0) et += NRELM1;                   // jnp negative-index wrap
        const float hn2 = sH[wave][lane * 16 + et];
        const float tn2 = sT[wave][lane * 16 + et];
        const float a   = hn2 * tn2;                // (hn*tn)^2 == hn^2 * tn^2
        att[e] = a;
        atomicMax(&segmax[head[e]], __float_as_uint(a));  // valid: a >= 0
    }
}

// ---------------------------------------------------------------------------
// Kernel 2: softmax numerator + segment denominator
// ---------------------------------------------------------------------------
__global__ __launch_bounds__(256)
void exp_sum_kernel(const int* __restrict__ head,
                    const float* __restrict__ segmax,
                    float* __restrict__ att,
                    float* __restrict__ segsum)
{
    const int e = blockIdx.x * 256 + threadIdx.x;
    if (e >= E_EDGES) return;
    const int h = head[e];
    const float ev = expf(att[e] - segmax[h]);
    att[e] = ev;                                    // overwrite in place with exp value
    atomicAdd(&segsum[h], ev);
}

// ---------------------------------------------------------------------------
// Kernel 3: entity aggregation from edges: agg_ent[head] += w * t * rel
// One wave per edge; lane handles 2 channels.
// ---------------------------------------------------------------------------
__global__ __launch_bounds__(256)
void edge_agg_kernel(const float* __restrict__ ent,
                     const float* __restrict__ weight,
                     const int* __restrict__ head,
                     const int* __restrict__ tail,
                     const int* __restrict__ etype,
                     const float* __restrict__ expv,
                     const float* __restrict__ segsum,
                     float* __restrict__ agg_ent)
{
    const int wave = threadIdx.x >> 5, lane = threadIdx.x & 31;
    const int e = blockIdx.x * 8 + wave;
    if (e >= E_EDGES) return;
    const int h = head[e], t = tail[e];
    int et = etype[e] - 1;
    if (et < 0) et += NRELM1;
    const float w = expv[e] / segsum[h];
    const int c = 2 * lane;
    const v2f tv = *(const v2f*)(ent + (size_t)t * CH + c);
    const v2f rv = *(const v2f*)(weight + et * CH + c);
    atomicAdd(agg_ent + (size_t)h * CH + c,     w * tv.x * rv.x);
    atomicAdd(agg_ent + (size_t)h * CH + c + 1, w * tv.y * rv.y);
}

// ---------------------------------------------------------------------------
// Kernel 4: bipartite SpMM both ways over interaction nnz. Wave per nnz.
// ---------------------------------------------------------------------------
__global__ __launch_bounds__(256)
void inter_agg_kernel(const float* __restrict__ cur_ent,
                      const float* __restrict__ cur_usr,
                      const float* __restrict__ vals,
                      const int* __restrict__ rows,
                      const int* __restrict__ cols,
                      float* __restrict__ agg_ent,
                      float* __restrict__ agg_usr)
{
    const int wave = threadIdx.x >> 5, lane = threadIdx.x & 31;
    const int z = blockIdx.x * 8 + wave;
    if (z >= NNZ_CNT) return;
    const int r = rows[z], cl = cols[z];
    const float v = vals[z];
    const int c = 2 * lane;
    const v2f uv = *(const v2f*)(cur_usr + (size_t)r  * CH + c);
    const v2f ev = *(const v2f*)(cur_ent + (size_t)cl * CH + c);
    atomicAdd(agg_ent + (size_t)cl * CH + c,     v * uv.x);
    atomicAdd(agg_ent + (size_t)cl * CH + c + 1, v * uv.y);
    atomicAdd(agg_usr + (size_t)r  * CH + c,     v * ev.x);
    atomicAdd(agg_usr + (size_t)r  * CH + c + 1, v * ev.y);
}

// ---------------------------------------------------------------------------
// Kernel 5: row-wise L2 normalize + residual accumulate. Wave per row (wave32).
// ---------------------------------------------------------------------------
__global__ __launch_bounds__(256)
void normalize_kernel(const float* __restrict__ agg,
                      float* __restrict__ cur,
                      float* __restrict__ out,
                      int nrows)
{
    const int wave = threadIdx.x >> 5, lane = threadIdx.x & 31;
    const int row = blockIdx.x * 8 + wave;
    if (row >= nrows) return;
    const int c = 2 * lane;
    v2f v = *(const v2f*)(agg + (size_t)row * CH + c);
    float ss = v.x * v.x + v.y * v.y;
#pragma unroll
    for (int m = 16; m >= 1; m >>= 1) ss += __shfl_xor(ss, m, 32);
    const float inv = 1.0f / fmaxf(sqrtf(ss), EPS_N);
    v.x *= inv; v.y *= inv;
    *(v2f*)(cur + (size_t)row * CH + c) = v;
    v2f o = *(v2f*)(out + (size_t)row * CH + c);
    o.x += v.x; o.y += v.y;
    *(v2f*)(out + (size_t)row * CH + c) = o;
}

// ---------------------------------------------------------------------------
extern "C" void kernel_launch(void* const* d_in, const int* in_sizes, int n_in,
                              void* d_out, int out_size, void* d_ws, size_t ws_size,
                              hipStream_t stream)
{
    (void)in_sizes; (void)n_in; (void)out_size; (void)ws_size;

    const float* user_emb   = (const float*)d_in[0];
    const float* entity_emb = (const float*)d_in[1];
    const float* weight     = (const float*)d_in[2];
    const float* inter_vals = (const float*)d_in[3];
    const int*   edge_index = (const int*)d_in[4];
    const int*   edge_type  = (const int*)d_in[5];
    const int*   inter_rows = (const int*)d_in[6];
    const int*   inter_cols = (const int*)d_in[7];
    const int*   head = edge_index;
    const int*   tail = edge_index + E_EDGES;

    float* out_ent = (float*)d_out;
    float* out_usr = out_ent + (size_t)N_ENT * CH;

    char* ws = (char*)d_ws;
    size_t off = 0;
    auto alloc_f = [&](size_t nfl) -> float* {
        float* p = (float*)(ws + off);
        off += ((nfl * sizeof(float) + 255) / 256) * 256;
        return p;
    };
    float* cur_ent = alloc_f((size_t)N_ENT * CH);
    float* cur_usr = alloc_f((size_t)N_USR * CH);
    float* agg_ent = alloc_f((size_t)N_ENT * CH);
    float* agg_usr = alloc_f((size_t)N_USR * CH);
    float* att     = alloc_f(E_EDGES);
    float* segmax  = alloc_f(N_ENT);
    float* segsum  = alloc_f(N_ENT);

    // init residual accumulators and current embeddings
    hipMemcpyAsync(cur_ent, entity_emb, (size_t)N_ENT * CH * sizeof(float), hipMemcpyDeviceToDevice, stream);
    hipMemcpyAsync(cur_usr, user_emb,   (size_t)N_USR * CH * sizeof(float), hipMemcpyDeviceToDevice, stream);
    hipMemcpyAsync(out_ent, entity_emb, (size_t)N_ENT * CH * sizeof(float), hipMemcpyDeviceToDevice, stream);
    hipMemcpyAsync(out_usr, user_emb,   (size_t)N_USR * CH * sizeof(float), hipMemcpyDeviceToDevice, stream);

    for (int hop = 0; hop < NHOPS; ++hop) {
        hipMemsetAsync(agg_ent, 0, (size_t)N_ENT * CH * sizeof(float), stream);
        hipMemsetAsync(agg_usr, 0, (size_t)N_USR * CH * sizeof(float), stream);
        hipMemsetAsync(segmax,  0, (size_t)N_ENT * sizeof(float), stream);
        hipMemsetAsync(segsum,  0, (size_t)N_ENT * sizeof(float), stream);

        att_wmma_kernel<<<E_EDGES / 16 / ATT_WAVES, ATT_WAVES * 32, 0, stream>>>(
            cur_ent, weight, head, tail, edge_type, att, (unsigned int*)segmax);
        exp_sum_kernel<<<(E_EDGES + 255) / 256, 256, 0, stream>>>(head, segmax, att, segsum);
        edge_agg_kernel<<<E_EDGES / 8, 256, 0, stream>>>(
            cur_ent, weight, head, tail, edge_type, att, segsum, agg_ent);
        inter_agg_kernel<<<NNZ_CNT / 8, 256, 0, stream>>>(
            cur_ent, cur_usr, inter_vals, inter_rows, inter_cols, agg_ent, agg_usr);
        normalize_kernel<<<(N_ENT + 7) / 8, 256, 0, stream>>>(agg_ent, cur_ent, out_ent, N_ENT);
        normalize_kernel<<<(N_USR + 7) / 8, 256, 0, stream>>>(agg_usr, cur_usr, out_usr, N_USR);
    }
}